// EuclideanVisitEncoder_69045894250727
// MI455X (gfx1250) — compile-verified
//
#include <hip/hip_runtime.h>
#include <stdint.h>

// EuclideanVisitEncoder: masked-mean embedding gather via the CDNA5 Tensor
// Data Mover in GATHER MODE.
//   code_ids : [N_VISITS, MAX_CODES] int32 (negative => padding)
//   emb      : [NUM_CODES, DIM]      float32   (100000 x 64, 256 B rows)
//   out      : [N_VISITS, DIM]       float32
//
// Roofline: 134 MFLOP vs ~264 MB moved; table (25.6 MB) is L2-resident on the
// 192 MB L2, so this is an L2-bandwidth gather problem. TDM gather mode
// (8 rows per descriptor, 32-bit indices) is the hardware embedding-lookup
// engine: row indices come straight from SGPRs (where the s_load'ed ids
// already live), the DMA engine lands 8 rows (2 KB) in LDS, the wave reduces
// from LDS with branchless mask-FMAs. Double-buffered, TENSORcnt-tracked.

#define NUM_CODES_T 100000
#define MAX_CODES_  64
#define DIM_        64
#define WAVES_PER_BLOCK 8
#define THREADS_PER_BLOCK (WAVES_PER_BLOCK * 32)
#define ROWS_PER_TDM 8
#define NCHUNK (MAX_CODES_ / ROWS_PER_TDM)   // 8 gather descriptors per visit

typedef uint32_t u32;
typedef __attribute__((ext_vector_type(4))) u32 u32x4;
typedef __attribute__((ext_vector_type(8))) u32 u32x8;

// tensor_load_to_lds with 4 descriptor groups in SGPRs (VIMAGE encoding):
//   g0: 4 SGPRs  (ctrl | lds_addr | global_addr | type)
//   g1: 8 SGPRs  (dims/strides/tile)
//   g2: 4 SGPRs  (gather row_index_0..3, 32-bit mode)
//   g3: 4 SGPRs  (gather row_index_4..7, 32-bit mode)
// Inline asm (not the clang builtin) so the source is portable across the
// ROCm 7.2 (5-arg builtin) and amdgpu-toolchain (6-arg builtin) frontends.
__device__ __forceinline__ void tdm_gather_rows(u32x4 g0, u32x8 g1,
                                                u32x4 g2, u32x4 g3) {
    // WAR guard: TDM is unordered vs DS ops; make sure any in-flight LDS
    // reads of the buffer we are about to overwrite have completed.
    asm volatile("s_wait_dscnt 0x0" ::: "memory");
    asm volatile("tensor_load_to_lds %0, %1, %2, %3"
                 :: "s"(g0), "s"(g1), "s"(g2), "s"(g3)
                 : "memory");
}

__global__ __launch_bounds__(THREADS_PER_BLOCK)
void visit_masked_mean_tdm(const int* __restrict__ code_ids,
                           const float* __restrict__ emb,
                           float* __restrict__ out,
                           int n_visits)
{
    // Per-wave double buffer: 2 x (8 rows x 64 f32) = 2 x 2 KB. 32 KB/block.
    __shared__ float stage[WAVES_PER_BLOCK][2][ROWS_PER_TDM * DIM_];

    const int lane = (int)(threadIdx.x & 31u);
    const int w = __builtin_amdgcn_readfirstlane((int)(threadIdx.x >> 5));
    const int n = __builtin_amdgcn_readfirstlane((int)(blockIdx.x * WAVES_PER_BLOCK) + w);
    if (n >= n_visits) return;

    const int* __restrict__ ids = code_ids + (size_t)n * MAX_CODES_; // uniform -> s_load

    // Flat address of a __shared__ object: low 32 bits == LDS byte offset
    const u32 lds_buf0 = (u32)(uintptr_t)&stage[w][0][0];
    const u32 lds_buf1 = (u32)(uintptr_t)&stage[w][1][0];

    const uint64_t gbase = (uint64_t)(uintptr_t)emb;

    // ---- D# group0 template (ctrl / lds / global address) ----
    u32x4 g0t;
    g0t[0] = 0xC0000001u;                 // gather_mode=1 | idx32=1 | count=1
    g0t[1] = 0u;                          // lds_addr, patched per issue
    g0t[2] = (u32)gbase;                  // global_addr[31:0] (table base)
    g0t[3] = ((u32)(gbase >> 32) & 0x01FFFFFFu) | (2u << 30); // addr[56:32], type=2

    // ---- D# group1 (geometry, constant) ----
    u32x8 g1;
    g1[0] = (2u << 16);                                   // data_size=2 -> 4 B
    g1[1] = ((u32)DIM_ & 0xFFFFu) << 16;                  // tensor_dim0[15:0]=64
    g1[2] = (((u32)DIM_ >> 16) & 0xFFFFu)
          | (((u32)NUM_CODES_T & 0xFFFFu) << 16);         // tensor_dim1 lo
    g1[3] = (((u32)NUM_CODES_T >> 16) & 0xFFFFu)
          | ((u32)DIM_ << 16);                            // dim1 hi | tile_dim0=64
    g1[4] = (u32)ROWS_PER_TDM;                            // tile_dim1 = 8 indices
    g1[5] = (u32)DIM_;                                    // tensor_dim0_stride = 64
    g1[6] = 0u;
    g1[7] = 0u;

    const int dpair = lane << 1;   // each lane owns dims [2*lane, 2*lane+1]

    float2 acc[ROWS_PER_TDM];
#pragma unroll
    for (int r = 0; r < ROWS_PER_TDM; ++r) { acc[r].x = 0.0f; acc[r].y = 0.0f; }
    int count = 0;

    // Issue chunk 0.
    {
        u32x4 g2, g3;
#pragma unroll
        for (int k = 0; k < 4; ++k) {
            const int a = ids[k], b = ids[4 + k];
            g2[k] = (u32)__builtin_amdgcn_readfirstlane((a < 0) ? 0 : a);
            g3[k] = (u32)__builtin_amdgcn_readfirstlane((b < 0) ? 0 : b);
        }
        u32x4 g0 = g0t; g0[1] = lds_buf0;
        tdm_gather_rows(g0, g1, g2, g3);
    }

#pragma unroll
    for (int chunk = 0; chunk < NCHUNK; ++chunk) {
        // Kick off the next DMA gather while we reduce the current buffer.
        if (chunk + 1 < NCHUNK) {
            const int base = (chunk + 1) * ROWS_PER_TDM;
            u32x4 g2, g3;
#pragma unroll
            for (int k = 0; k < 4; ++k) {
                const int a = ids[base + k], b = ids[base + 4 + k];
                g2[k] = (u32)__builtin_amdgcn_readfirstlane((a < 0) ? 0 : a);
                g3[k] = (u32)__builtin_amdgcn_readfirstlane((b < 0) ? 0 : b);
            }
            u32x4 g0 = g0t; g0[1] = ((chunk + 1) & 1) ? lds_buf1 : lds_buf0;
            tdm_gather_rows(g0, g1, g2, g3);
            __builtin_amdgcn_s_wait_tensorcnt(1);   // current chunk's DMA done
        } else {
            __builtin_amdgcn_s_wait_tensorcnt(0);   // last chunk: drain
        }
        asm volatile("" ::: "memory");  // pin the LDS reads behind the wait

        const float* buf = &stage[w][chunk & 1][0];
        const int base = chunk * ROWS_PER_TDM;
#pragma unroll
        for (int r = 0; r < ROWS_PER_TDM; ++r) {
            const int idr = ids[base + r];
            const float m = (idr >= 0) ? 1.0f : 0.0f;    // branchless mask
            count += (idr >= 0) ? 1 : 0;
            const float2 v = *reinterpret_cast<const float2*>(buf + r * DIM_ + dpair);
            acc[r].x = fmaf(v.x, m, acc[r].x);
            acc[r].y = fmaf(v.y, m, acc[r].y);
        }
    }

    float sx = ((acc[0].x + acc[1].x) + (acc[2].x + acc[3].x)) +
               ((acc[4].x + acc[5].x) + (acc[6].x + acc[7].x));
    float sy = ((acc[0].y + acc[1].y) + (acc[2].y + acc[3].y)) +
               ((acc[4].y + acc[5].y) + (acc[6].y + acc[7].y));

    // mean over valid ids; count==0 -> sums are 0 -> 0/1 = 0 (matches where()).
    const float denom = (float)((count > 0) ? count : 1);
    float2 res;
    res.x = sx / denom;
    res.y = sy / denom;

    *reinterpret_cast<float2*>(out + (size_t)n * DIM_ + dpair) = res;
}

extern "C" void kernel_launch(void* const* d_in, const int* in_sizes, int n_in,
                              void* d_out, int out_size, void* d_ws, size_t ws_size,
                              hipStream_t stream) {
    const int*   code_ids = (const int*)d_in[0];   // [N, 64] int32
    const float* emb      = (const float*)d_in[1]; // [100000, 64] f32
    float*       out      = (float*)d_out;         // [N, 64] f32

    const int n_visits = in_sizes[0] / MAX_CODES_;
    const int blocks = (n_visits + WAVES_PER_BLOCK - 1) / WAVES_PER_BLOCK;

    hipLaunchKernelGGL(visit_masked_mean_tdm,
                       dim3(blocks), dim3(THREADS_PER_BLOCK), 0, stream,
                       code_ids, emb, out, n_visits);
}